// QuantumEcologicalModel_15418932592872
// MI455X (gfx1250) — compile-verified
//
#include <hip/hip_runtime.h>
#include <cstdint>
#include <cmath>

#define NQ 20
#define MS 8
#define DIMN (1u << NQ)   // 2^20
#define H1D 128
#define H2D 64

typedef float v2f __attribute__((ext_vector_type(2)));
typedef float v8f __attribute__((ext_vector_type(8)));

// ---------------------------------------------------------------------------
// CNOT ladder: composed source index i -> src_G1(src_G2(...src_Gk(i))).
// We need bit19 of that composed source as a GF(2) linear form parity(i&mask).
// Process gates in application order (i ascending, j ascending), substituting
// x_{j-1} = src_{Gj}(x_j): mask' = mask ^ (bit_t(mask) ? (1<<c) : 0).
// ---------------------------------------------------------------------------
static constexpr unsigned cnot_mask() {
  unsigned m = 1u << (NQ - 1);
  for (int c = 0; c < NQ; ++c)
    for (int t = c + 1; t < NQ; ++t)
      if ((m >> t) & 1u) m ^= (1u << c);
  return m;
}
static constexpr unsigned PAR_MASK = cnot_mask();

// ---------------------------------------------------------------------------
// Kernel A: scalar emulation of the collapsed state. After every buggy diag
// gate the state is a function of one bit only -> track (g0, g1).
//   cross-bit rule:  S[b] = 2^(N-2)*(g0+g1);  g'[b] = diag[b]*S
//   same-bit rule:   S[b] = 2^(N-1)*g[b];     g'[b] = diag[b]*S[b]
// ---------------------------------------------------------------------------
__global__ void k_prep(const float* __restrict__ theta, float* __restrict__ pw) {
  if (threadIdx.x != 0) return;
  const float A  = 0.70710678118654752440f;  // fl(1/sqrt(2))
  const float F2 = 262144.0f;                // 2^(N-2)
  const float F1 = 524288.0f;                // 2^(N-1)
  // First gate (H on qubit 0) on e0: S = [1, 0]
  float g0 = A, g1 = 0.0f;
  int prev = 0;
  // Remaining Hadamards, qubits 1..19 (always cross-bit)
  for (int q = 1; q < NQ; ++q) {
    float s = (g0 + g1) * F2;
    g0 = A * s;
    g1 = -A * s;   // exact negation of g0 -> exact cancellation next gate
    prev = q;
  }
  // RX diag gates: diag = [cos(theta/2), cos(theta/2)], 8 per qubit
  for (int i = 0; i < NQ; ++i) {
    for (int j = 0; j < MS; ++j) {
      float c = cosf(theta[i * MS + j] * 0.5f);
      if (i == prev) {
        g0 = c * (g0 * F1);
        g1 = c * (g1 * F1);
      } else {
        float s = (g0 + g1) * F2;
        g0 = c * s;
        g1 = c * s;
        prev = i;
      }
    }
  }
  pw[0] = g0 * g0;  // prob value for parity-0 indices
  pw[1] = g1 * g1;  // prob value for parity-1 indices
}

// ---------------------------------------------------------------------------
// Kernel B: parity-split column sums of W1 (DIM x 128), full-bandwidth path.
// Each wave streams whole 512B rows (float4 per lane, global_load_b128).
// Exact early-skip when p0==p1==0 (their contribution is identically zero).
// Deterministic: fixed per-wave loop order, fixed-order LDS reduce, per-block
// partials reduced in fixed order by kernel C. No float atomics.
// ---------------------------------------------------------------------------
__global__ __launch_bounds__(256) void k_colsum(const float* __restrict__ W1,
                                                const float* __restrict__ pw,
                                                float* __restrict__ partial,
                                                int nblocks) {
  __shared__ float sh[8 * 256];
  const int wave = threadIdx.x >> 5;
  const int lane = threadIdx.x & 31;
  float a0[4] = {0.f, 0.f, 0.f, 0.f};
  float a1[4] = {0.f, 0.f, 0.f, 0.f};

  const bool zero = (pw[0] == 0.0f) && (pw[1] == 0.0f);
  if (!zero) {
    const unsigned gw = (unsigned)blockIdx.x * 8u + (unsigned)wave;
    const unsigned nw = (unsigned)nblocks * 8u;
    for (unsigned r = gw; r < DIMN; r += nw) {
      const float4 v =
          *reinterpret_cast<const float4*>(W1 + (size_t)r * H1D + lane * 4);
      if (__popc(r & PAR_MASK) & 1) {
        a1[0] += v.x; a1[1] += v.y; a1[2] += v.z; a1[3] += v.w;
      } else {
        a0[0] += v.x; a0[1] += v.y; a0[2] += v.z; a0[3] += v.w;
      }
    }
  }
#pragma unroll
  for (int q = 0; q < 4; ++q) {
    sh[wave * 256 + 0 * 128 + lane * 4 + q] = a0[q];
    sh[wave * 256 + 1 * 128 + lane * 4 + q] = a1[q];
  }
  __syncthreads();
  const int s = threadIdx.x;  // slot: [0,128) = class0 cols, [128,256) = class1
  float v = 0.f;
#pragma unroll
  for (int w = 0; w < 8; ++w) v += sh[w * 256 + s];
  partial[(size_t)blockIdx.x * 256 + s] = v;
}

// ---------------------------------------------------------------------------
// Kernel C: finish reduction -> h1 = relu(p0*c0 + p1*c1 + b1), then the
// 128->64->20 MLP via V_WMMA_F32_16X16X4_F32 (wave 0 only; M=16 tile, row 0
// carries the live vector; full f32 precision).
//   A 16x4 layout:  lanes 0-15 = M rows, v0=K0/K2, v1=K1/K3 (lane<16 / >=16)
//   B 4x16 layout:  lanes = N cols, v0=K0/K2, v1=K1/K3
//   C 16x16 layout: VGPR0 lanes 0-15 -> M=0 row, N=lane
// ---------------------------------------------------------------------------
__global__ __launch_bounds__(128) void k_mlp(const float* __restrict__ pw,
                                             const float* __restrict__ partial,
                                             int nblocks,
                                             const float* __restrict__ b1,
                                             const float* __restrict__ W2,
                                             const float* __restrict__ b2,
                                             const float* __restrict__ W3,
                                             const float* __restrict__ b3,
                                             float* __restrict__ out) {
  __shared__ float h1s[H1D];
  __shared__ float h2s[H2D];
  const int t = threadIdx.x;

  if (t < H1D) {
    float c0 = 0.f, c1 = 0.f;
    for (int b = 0; b < nblocks; ++b) {
      c0 += partial[(size_t)b * 256 + t];
      c1 += partial[(size_t)b * 256 + 128 + t];
    }
    float y = pw[0] * c0 + pw[1] * c1 + b1[t];
    h1s[t] = y > 0.f ? y : 0.f;
  }
  __syncthreads();

  const int lane = t;          // wave 0 lanes 0..31
  const bool lo = lane < 16;
  const int l16 = lane & 15;

  // ---- layer 2: h2 = relu(h1 @ W2 + b2), W2 is 128x64 row-major ----
  if (t < 32) {
#pragma unroll
    for (int nt = 0; nt < 4; ++nt) {
      v8f c = {0.f, 0.f, 0.f, 0.f, 0.f, 0.f, 0.f, 0.f};
      const int col = nt * 16 + l16;
      for (int k0 = 0; k0 < H1D; k0 += 4) {
        v2f a, b;
        a[0] = (lane == 0) ? h1s[k0 + 0] : ((lane == 16) ? h1s[k0 + 2] : 0.f);
        a[1] = (lane == 0) ? h1s[k0 + 1] : ((lane == 16) ? h1s[k0 + 3] : 0.f);
        const int kk = lo ? k0 : k0 + 2;
        b[0] = W2[(size_t)(kk + 0) * H2D + col];
        b[1] = W2[(size_t)(kk + 1) * H2D + col];
        c = __builtin_amdgcn_wmma_f32_16x16x4_f32(false, a, false, b,
                                                  (short)0, c, false, false);
      }
      if (lo) {
        float y = c[0] + b2[col];  // C row M=0: VGPR0, lanes 0..15, N=lane
        h2s[col] = y > 0.f ? y : 0.f;
      }
    }
  }
  __syncthreads();

  // ---- layer 3: out = h2 @ W3 + b3, W3 is 64x20 row-major ----
  if (t < 32) {
#pragma unroll
    for (int nt = 0; nt < 2; ++nt) {
      v8f c = {0.f, 0.f, 0.f, 0.f, 0.f, 0.f, 0.f, 0.f};
      const int col = nt * 16 + l16;
      const bool valid = col < NQ;
      for (int k0 = 0; k0 < H2D; k0 += 4) {
        v2f a, b;
        a[0] = (lane == 0) ? h2s[k0 + 0] : ((lane == 16) ? h2s[k0 + 2] : 0.f);
        a[1] = (lane == 0) ? h2s[k0 + 1] : ((lane == 16) ? h2s[k0 + 3] : 0.f);
        const int kk = lo ? k0 : k0 + 2;
        b[0] = valid ? W3[(size_t)(kk + 0) * NQ + col] : 0.f;
        b[1] = valid ? W3[(size_t)(kk + 1) * NQ + col] : 0.f;
        c = __builtin_amdgcn_wmma_f32_16x16x4_f32(false, a, false, b,
                                                  (short)0, c, false, false);
      }
      if (lo && col < NQ) out[col] = c[0] + b3[col];
    }
  }
}

// ---------------------------------------------------------------------------
extern "C" void kernel_launch(void* const* d_in, const int* in_sizes, int n_in,
                              void* d_out, int out_size, void* d_ws,
                              size_t ws_size, hipStream_t stream) {
  (void)in_sizes; (void)n_in; (void)out_size;
  // setup_inputs order: x, theta, W1, b1, W2, b2, W3, b3 (all float32)
  const float* theta = (const float*)d_in[1];
  const float* W1    = (const float*)d_in[2];
  const float* b1    = (const float*)d_in[3];
  const float* W2    = (const float*)d_in[4];
  const float* b2    = (const float*)d_in[5];
  const float* W3    = (const float*)d_in[6];
  const float* b3    = (const float*)d_in[7];
  float* out = (float*)d_out;

  float* pw      = (float*)d_ws;   // pw[0]=p0, pw[1]=p1
  float* partial = pw + 64;        // 256B-aligned partial buffer

  long avail = (long)(ws_size / 4) - 64;
  int nb = (int)(avail / 256);     // blocks we have partial space for
  if (nb > 2048) nb = 2048;        // 2048 blocks x 256 thr saturates HBM
  if (nb < 1) nb = 1;

  k_prep  <<<1, 32, 0, stream>>>(theta, pw);
  k_colsum<<<nb, 256, 0, stream>>>(W1, pw, partial, nb);
  k_mlp   <<<1, 128, 0, stream>>>(pw, partial, nb, b1, W2, b2, W3, b3, out);
}